// GraphNet_25769803996
// MI455X (gfx1250) — compile-verified
//
#include <hip/hip_runtime.h>

typedef __attribute__((ext_vector_type(2))) float v2f;
typedef __attribute__((ext_vector_type(8))) float v8f;

#define LDH 52  // LDS tile row stride in floats: conflict-free & 16B aligned

__device__ __forceinline__ v8f zero8() {
  v8f z = {0.f, 0.f, 0.f, 0.f, 0.f, 0.f, 0.f, 0.f};
  return z;
}

// A-fragment: lane holds rows `col`, K-pair 2*hl .. 2*hl+1 of the k-step
__device__ __forceinline__ v2f lda(const float* hs, int k, int hl, int col) {
  v2f a;
  const float* ap = hs + col * LDH + k + 2 * hl;
  a[0] = ap[0];
  a[1] = ap[1];
  return a;
}

// Register-resident weights for one 4-layer MLP (B fragments + biases + LN)
template <int KIN>
struct WReg {
  v2f w0[KIN / 4];
  v2f w1[4], w2[4], w3[4];
  float b0, b1, b2, b3, G, Bt;
};

template <int KIN, bool LN>
__device__ __forceinline__ void load_wreg(WReg<KIN>& R, const float* wl,
                                          int hl, int col) {
  const float* Win = wl;
  const float* bin = Win + KIN * 16;
  const float* Wh0 = bin + 16;
  const float* bh0 = Wh0 + 256;
  const float* Wh1 = bh0 + 16;
  const float* bh1 = Wh1 + 256;
  const float* Wout = bh1 + 16;
  const float* bout = Wout + 256;
#pragma unroll
  for (int i = 0; i < KIN / 4; ++i) {
    const float* bp = Win + (4 * i + 2 * hl) * 16 + col;
    R.w0[i][0] = bp[0];
    R.w0[i][1] = bp[16];
  }
#pragma unroll
  for (int i = 0; i < 4; ++i) {
    const float* bp = Wh0 + (4 * i + 2 * hl) * 16 + col;
    R.w1[i][0] = bp[0];
    R.w1[i][1] = bp[16];
  }
#pragma unroll
  for (int i = 0; i < 4; ++i) {
    const float* bp = Wh1 + (4 * i + 2 * hl) * 16 + col;
    R.w2[i][0] = bp[0];
    R.w2[i][1] = bp[16];
  }
#pragma unroll
  for (int i = 0; i < 4; ++i) {
    const float* bp = Wout + (4 * i + 2 * hl) * 16 + col;
    R.w3[i][0] = bp[0];
    R.w3[i][1] = bp[16];
  }
  R.b0 = bin[col];
  R.b1 = bh0[col];
  R.b2 = bh1[col];
  R.b3 = bout[col];
  if (LN) {
    R.G = bout[16 + col];
    R.Bt = bout[32 + col];
  } else {
    R.G = 1.f;
    R.Bt = 0.f;
  }
}

__device__ __forceinline__ void relu_store(float* hs, v8f c, float b, int hl,
                                           int col) {
#pragma unroll
  for (int g = 0; g < 8; ++g)
    hs[(g + 8 * hl) * LDH + col] = fmaxf(c[g] + b, 0.f);
}

// LayerNorm over 16 features (each C/D VGPR is one row across a 16-lane half)
__device__ __forceinline__ v8f ln_apply(v8f c, float b, float G, float Bt) {
  v8f o;
#pragma unroll
  for (int g = 0; g < 8; ++g) {
    float h = c[g] + b;
    float s = h;
    s += __shfl_xor(s, 1, 32);
    s += __shfl_xor(s, 2, 32);
    s += __shfl_xor(s, 4, 32);
    s += __shfl_xor(s, 8, 32);
    float mu = s * 0.0625f;
    float d = h - mu;
    float q = d * d;
    q += __shfl_xor(q, 1, 32);
    q += __shfl_xor(q, 2, 32);
    q += __shfl_xor(q, 4, 32);
    q += __shfl_xor(q, 8, 32);
    o[g] = d * rsqrtf(q * 0.0625f + 1e-5f) * G + Bt;
  }
  return o;
}

// 4-layer MLP on the 16-row LDS tile, weights in registers
template <int KIN, bool LN>
__device__ __forceinline__ v8f mlp_tile(float* hs, const WReg<KIN>& R, int hl,
                                        int col) {
  v8f c = zero8();
#pragma unroll
  for (int i = 0; i < KIN / 4; ++i) {
    v2f a = lda(hs, 4 * i, hl, col);
    c = __builtin_amdgcn_wmma_f32_16x16x4_f32(false, a, false, R.w0[i],
                                              (short)0, c, false, false);
  }
  relu_store(hs, c, R.b0, hl, col);
  c = zero8();
#pragma unroll
  for (int i = 0; i < 4; ++i) {
    v2f a = lda(hs, 4 * i, hl, col);
    c = __builtin_amdgcn_wmma_f32_16x16x4_f32(false, a, false, R.w1[i],
                                              (short)0, c, false, false);
  }
  relu_store(hs, c, R.b1, hl, col);
  c = zero8();
#pragma unroll
  for (int i = 0; i < 4; ++i) {
    v2f a = lda(hs, 4 * i, hl, col);
    c = __builtin_amdgcn_wmma_f32_16x16x4_f32(false, a, false, R.w2[i],
                                              (short)0, c, false, false);
  }
  relu_store(hs, c, R.b2, hl, col);
  c = zero8();
#pragma unroll
  for (int i = 0; i < 4; ++i) {
    v2f a = lda(hs, 4 * i, hl, col);
    c = __builtin_amdgcn_wmma_f32_16x16x4_f32(false, a, false, R.w3[i],
                                              (short)0, c, false, false);
  }
  if (LN) {
    return ln_apply(c, R.b3, R.G, R.Bt);
  } else {
    v8f o;
#pragma unroll
    for (int g = 0; g < 8; ++g) o[g] = c[g] + R.b3;
    return o;
  }
}

struct MLPW {
  const float *Win, *bin, *Wh, *bh, *Wout, *bout, *g, *beta;
};

__device__ void stage_mlp(float* wl, MLPW w, int KIN, bool ln) {
  const int t = threadIdx.x, B = blockDim.x;
  int o = 0;
  for (int k = t; k < KIN * 16; k += B) wl[k] = w.Win[k];
  o = KIN * 16;
  for (int k = t; k < 16; k += B) wl[o + k] = w.bin[k];
  o += 16;
  for (int k = t; k < 256; k += B) wl[o + k] = w.Wh[k];
  o += 256;
  for (int k = t; k < 16; k += B) wl[o + k] = w.bh[k];
  o += 16;
  for (int k = t; k < 256; k += B) wl[o + k] = w.Wh[256 + k];
  o += 256;
  for (int k = t; k < 16; k += B) wl[o + k] = w.bh[16 + k];
  o += 16;
  for (int k = t; k < 256; k += B) wl[o + k] = w.Wout[k];
  o += 256;
  for (int k = t; k < 16; k += B) wl[o + k] = w.bout[k];
  o += 16;
  if (ln) {
    for (int k = t; k < 16; k += B) wl[o + k] = w.g[k];
    o += 16;
    for (int k = t; k < 16; k += B) wl[o + k] = w.beta[k];
  }
}

__global__ void __launch_bounds__(256) zero_kernel(float4* p, int n4) {
  int i = blockIdx.x * blockDim.x + threadIdx.x;
  if (i < n4) p[i] = make_float4(0.f, 0.f, 0.f, 0.f);
}

// ---- node encoder: x[N,12] -> hn[N,16]
__global__ void __launch_bounds__(256)
enc_node_kernel(const float* __restrict__ x, float* __restrict__ hn, MLPW w,
                int ntile) {
  __shared__ float tiles[8 * 16 * LDH];
  __shared__ float wl[12 * 16 + 864];
  stage_mlp(wl, w, 12, true);
  __syncthreads();
  const int lane = threadIdx.x & 31, warp = threadIdx.x >> 5;
  const int hl = lane >> 4, col = lane & 15;
  float* hs = tiles + warp * 16 * LDH;
  WReg<12> R;
  load_wreg<12, true>(R, wl, hl, col);
  for (int tile = blockIdx.x * 8 + warp; tile < ntile;
       tile += gridDim.x * 8) {
    const int n0 = tile * 16;
    if (lane < 16) {
      const float4* xr = (const float4*)(x + (size_t)(n0 + col) * 12);
      float4* hr = (float4*)(hs + col * LDH);
      hr[0] = xr[0];
      hr[1] = xr[1];
      hr[2] = xr[2];
    }
    v8f o = mlp_tile<12, true>(hs, R, hl, col);
#pragma unroll
    for (int g = 0; g < 8; ++g)
      hn[(size_t)(n0 + g + 8 * hl) * 16 + col] = o[g];
  }
}

// ---- edge encoder: dist[E,4] -> he[E,16]
__global__ void __launch_bounds__(256)
enc_edge_kernel(const float* __restrict__ dist, float* __restrict__ he, MLPW w,
                int ntile) {
  __shared__ float tiles[8 * 16 * LDH];
  __shared__ float wl[4 * 16 + 864];
  stage_mlp(wl, w, 4, true);
  __syncthreads();
  const int lane = threadIdx.x & 31, warp = threadIdx.x >> 5;
  const int hl = lane >> 4, col = lane & 15;
  float* hs = tiles + warp * 16 * LDH;
  WReg<4> R;
  load_wreg<4, true>(R, wl, hl, col);
  for (int tile = blockIdx.x * 8 + warp; tile < ntile;
       tile += gridDim.x * 8) {
    const int e0 = tile * 16;
    if (lane < 16) {
      const float4* dr = (const float4*)(dist + (size_t)(e0 + col) * 4);
      float4* hr = (float4*)(hs + col * LDH);
      hr[0] = dr[0];
    }
    v8f o = mlp_tile<4, true>(hs, R, hl, col);
#pragma unroll
    for (int g = 0; g < 8; ++g)
      he[(size_t)(e0 + g + 8 * hl) * 16 + col] = o[g];
  }
}

// ---- processor edge step: gather hn[src],hn[dst]; he += MLP([he|hs|hd]);
//      atomic scatter-add of updated he into pe_sum[dst]
__global__ void __launch_bounds__(256)
edge_kernel(float* __restrict__ he, const float* __restrict__ hn,
            float* __restrict__ pes, const int* __restrict__ src,
            const int* __restrict__ dstI, MLPW w, int ntile) {
  __shared__ float tiles[8 * 16 * LDH];
  __shared__ int sidx[8 * 32];
  __shared__ float wl[48 * 16 + 864];
  stage_mlp(wl, w, 48, true);
  __syncthreads();
  const int lane = threadIdx.x & 31, warp = threadIdx.x >> 5;
  const int hl = lane >> 4, col = lane & 15;
  float* hs = tiles + warp * 16 * LDH;
  int* idx = sidx + warp * 32;
  WReg<48> R;
  load_wreg<48, true>(R, wl, hl, col);
  for (int tile = blockIdx.x * 8 + warp; tile < ntile;
       tile += gridDim.x * 8) {
    const int e0 = tile * 16;
    {  // he rows -> tile cols 0..15 (each lane: 2 float4 of row `col`)
      const float4* er = (const float4*)(he + (size_t)(e0 + col) * 16);
      float4* hr = (float4*)(hs + col * LDH);
      hr[2 * hl] = er[2 * hl];
      hr[2 * hl + 1] = er[2 * hl + 1];
    }
    idx[lane] = hl ? dstI[e0 + col] : src[e0 + col];  // [0..15]=src [16..31]=dst
    {  // gather hn rows: src -> cols 16..31, dst -> cols 32..47
      int gi = idx[lane];
      const float4* gr = (const float4*)(hn + (size_t)gi * 16);
      float4* hr = (float4*)(hs + col * LDH + 16 + 16 * hl);
      hr[0] = gr[0];
      hr[1] = gr[1];
      hr[2] = gr[2];
      hr[3] = gr[3];
    }
    float res[8];  // residual copy of he before the tile gets overwritten
#pragma unroll
    for (int g = 0; g < 8; ++g) res[g] = hs[(g + 8 * hl) * LDH + col];

    v8f o = mlp_tile<48, true>(hs, R, hl, col);

#pragma unroll
    for (int g = 0; g < 8; ++g) {
      int m = g + 8 * hl;
      float nh = res[g] + o[g];
      he[(size_t)(e0 + m) * 16 + col] = nh;
      int di = idx[16 + m];
      atomicAdd(&pes[(size_t)di * 16 + col], nh);
    }
  }
}

// ---- processor node step: hn += MLP([hn | pe_sum])
__global__ void __launch_bounds__(256)
node_kernel(float* __restrict__ hn, const float* __restrict__ pes, MLPW w,
            int ntile) {
  __shared__ float tiles[8 * 16 * LDH];
  __shared__ float wl[32 * 16 + 864];
  stage_mlp(wl, w, 32, true);
  __syncthreads();
  const int lane = threadIdx.x & 31, warp = threadIdx.x >> 5;
  const int hl = lane >> 4, col = lane & 15;
  float* hs = tiles + warp * 16 * LDH;
  WReg<32> R;
  load_wreg<32, true>(R, wl, hl, col);
  for (int tile = blockIdx.x * 8 + warp; tile < ntile;
       tile += gridDim.x * 8) {
    const int n0 = tile * 16;
    {  // lanes 0..15: hn row -> cols 0..15 ; lanes 16..31: pe_sum -> 16..31
      const float* sp = hl ? (pes + (size_t)(n0 + col) * 16)
                           : (hn + (size_t)(n0 + col) * 16);
      const float4* sr = (const float4*)sp;
      float4* hr = (float4*)(hs + col * LDH + 16 * hl);
      hr[0] = sr[0];
      hr[1] = sr[1];
      hr[2] = sr[2];
      hr[3] = sr[3];
    }
    float res[8];
#pragma unroll
    for (int g = 0; g < 8; ++g) res[g] = hs[(g + 8 * hl) * LDH + col];
    v8f o = mlp_tile<32, true>(hs, R, hl, col);
#pragma unroll
    for (int g = 0; g < 8; ++g)
      hn[(size_t)(n0 + g + 8 * hl) * 16 + col] = res[g] + o[g];
  }
}

// ---- decoder: hn[N,16] -> out[N,4] (weights zero-padded 4 -> 16 wide)
__global__ void __launch_bounds__(256)
dec_kernel(const float* __restrict__ hn, float* __restrict__ out, MLPW w,
           int ntile) {
  __shared__ float tiles[8 * 16 * LDH];
  __shared__ float wl[256 + 832];
  {
    const int t = threadIdx.x, B = blockDim.x;
    for (int k = t; k < 256 + 832; k += B) wl[k] = 0.f;
    __syncthreads();
    for (int k = t; k < 64; k += B) {
      int r = k >> 2, c2 = k & 3;
      wl[r * 16 + c2] = w.Win[k];
    }
    int o = 256;
    for (int k = t; k < 4; k += B) wl[o + k] = w.bin[k];
    o += 16;
    for (int k = t; k < 16; k += B) {
      int r = k >> 2, c2 = k & 3;
      wl[o + r * 16 + c2] = w.Wh[k];
    }
    o += 256;
    for (int k = t; k < 4; k += B) wl[o + k] = w.bh[k];
    o += 16;
    for (int k = t; k < 16; k += B) {
      int r = k >> 2, c2 = k & 3;
      wl[o + r * 16 + c2] = w.Wh[16 + k];
    }
    o += 256;
    for (int k = t; k < 4; k += B) wl[o + k] = w.bh[4 + k];
    o += 16;
    for (int k = t; k < 16; k += B) {
      int r = k >> 2, c2 = k & 3;
      wl[o + r * 16 + c2] = w.Wout[k];
    }
    o += 256;
    for (int k = t; k < 4; k += B) wl[o + k] = w.bout[k];
  }
  __syncthreads();
  const int lane = threadIdx.x & 31, warp = threadIdx.x >> 5;
  const int hl = lane >> 4, col = lane & 15;
  float* hs = tiles + warp * 16 * LDH;
  WReg<16> R;
  load_wreg<16, false>(R, wl, hl, col);
  for (int tile = blockIdx.x * 8 + warp; tile < ntile;
       tile += gridDim.x * 8) {
    const int n0 = tile * 16;
    {
      const float4* sr = (const float4*)(hn + (size_t)(n0 + col) * 16);
      float4* hr = (float4*)(hs + col * LDH);
      hr[2 * hl] = sr[2 * hl];
      hr[2 * hl + 1] = sr[2 * hl + 1];
    }
    v8f o = mlp_tile<16, false>(hs, R, hl, col);
#pragma unroll
    for (int g = 0; g < 8; ++g)
      if (col < 4) out[(size_t)(n0 + g + 8 * hl) * 4 + col] = o[g];
  }
}

extern "C" void kernel_launch(void* const* d_in, const int* in_sizes, int n_in,
                              void* d_out, int out_size, void* d_ws,
                              size_t ws_size, hipStream_t stream) {
  (void)in_sizes; (void)n_in; (void)out_size; (void)ws_size;
  const float* x = (const float*)d_in[0];
  const float* dist = (const float*)d_in[1];
  const int* src = (const int*)d_in[2];
  const int* dstI = (const int*)d_in[3];
  auto F = [&](int i) { return (const float*)d_in[i]; };
  MLPW enn{F(4), F(5), F(6), F(7), F(8), F(9), F(10), F(11)};
  MLPW ene{F(12), F(13), F(14), F(15), F(16), F(17), F(18), F(19)};
  MLPW dcw{F(36), F(37), F(38), F(39), F(40), F(41), nullptr, nullptr};

  const int N = 100000, E = 3200000;
  float* ws = (float*)d_ws;
  float* hn = ws;                    // [N,16]
  float* pes = ws + (size_t)N * 16;  // [N,16]
  float* he = ws + (size_t)N * 32;   // [E,16]

  const int ntileN = N / 16;   // 6250
  const int ntileE = E / 16;   // 200000
  const int gridE = 1024;      // persistent: 8192 waves, ~24 tiles each
  const int gridN = 391;       // persistent: 3128 waves, ~2 tiles each

  enc_node_kernel<<<gridN, 256, 0, stream>>>(x, hn, enn, ntileN);
  enc_edge_kernel<<<gridE, 256, 0, stream>>>(dist, he, ene, ntileE);

  for (int i = 0; i < 4; ++i) {
    MLPW pe{F(20) + i * 768, F(21) + i * 16, F(22) + i * 512, F(23) + i * 32,
            F(24) + i * 256, F(25) + i * 16, F(26) + i * 16, F(27) + i * 16};
    MLPW pn{F(28) + i * 512, F(29) + i * 16, F(30) + i * 512, F(31) + i * 32,
            F(32) + i * 256, F(33) + i * 16, F(34) + i * 16, F(35) + i * 16};
    zero_kernel<<<(N * 4 + 255) / 256, 256, 0, stream>>>((float4*)pes, N * 4);
    edge_kernel<<<gridE, 256, 0, stream>>>(he, hn, pes, src, dstI, pe, ntileE);
    node_kernel<<<gridN, 256, 0, stream>>>(hn, pes, pn, ntileN);
  }
  dec_kernel<<<gridN, 256, 0, stream>>>(hn, (float*)d_out, dcw, ntileN);
}